// NATLayer_76261439308281
// MI455X (gfx1250) — compile-verified
//
#include <hip/hip_runtime.h>
#include <math.h>

typedef __attribute__((ext_vector_type(2))) float v2f;
typedef __attribute__((ext_vector_type(8))) float v8f;

#define Bb   4
#define Hh   64
#define Ww   64
#define Cc   128
#define NHh  4
#define HDd  32
#define Kk   7
#define Mtok (Bb*Hh*Ww)   /* 16384 tokens */
#define MLPD_ 512

// ---------------------------------------------------------------------------
// LayerNorm: one wave32 per token (C=128 -> float4 per lane), 8 waves/block.
// ---------------------------------------------------------------------------
__global__ __launch_bounds__(256) void ln_kernel(const float* __restrict__ x,
                                                 const float* __restrict__ g,
                                                 const float* __restrict__ b,
                                                 float* __restrict__ y, int M) {
  int wid = blockIdx.x * 8 + (threadIdx.x >> 5);
  if (wid >= M) return;
  int lane = threadIdx.x & 31;
  const float4* xp = (const float4*)(x + (size_t)wid * Cc);
  float4 v = xp[lane];
  float s  = v.x + v.y + v.z + v.w;
  float s2 = v.x*v.x + v.y*v.y + v.z*v.z + v.w*v.w;
  #pragma unroll
  for (int off = 16; off; off >>= 1) {
    s  += __shfl_xor(s,  off, 32);
    s2 += __shfl_xor(s2, off, 32);
  }
  float mean = s * (1.0f / Cc);
  float var  = s2 * (1.0f / Cc) - mean * mean;
  float rstd = rsqrtf(var + 1e-5f);
  float4 gv = ((const float4*)g)[lane];
  float4 bv = ((const float4*)b)[lane];
  float4 o;
  o.x = (v.x - mean) * rstd * gv.x + bv.x;
  o.y = (v.y - mean) * rstd * gv.y + bv.y;
  o.z = (v.z - mean) * rstd * gv.z + bv.z;
  o.w = (v.w - mean) * rstd * gv.w + bv.w;
  ((float4*)(y + (size_t)wid * Cc))[lane] = o;
}

// ---------------------------------------------------------------------------
// FP32 WMMA GEMM: D = act((A @ Bw + bias) * alpha) + resid
//   A: MxKd  Bw: KdxN  bias: N  resid: MxN (or null).  Requires N % 64 == 0.
// One wave computes a 16x64 strip: 4 accumulators share one A fragment per
// K-step (4x fewer A fetches, 4 independent v_wmma chains to hide latency).
// V_WMMA_F32_16X16X4_F32 fragment layout:
//   A (16x4): lane<16 holds M=lane,K=k+{0,1}; lane>=16 holds K=k+{2,3}
//   B (4x16): same with N = lane&15
//   C/D (16x16): VGPR j -> row j (lanes 0-15) / row j+8 (lanes 16-31)
// ---------------------------------------------------------------------------
__global__ __launch_bounds__(256) void gemm_wmma(const float* __restrict__ A,
                                                 const float* __restrict__ Bw,
                                                 const float* __restrict__ bias,
                                                 const float* __restrict__ resid,
                                                 float* __restrict__ D,
                                                 int M, int N, int Kd,
                                                 float alpha, int act) {
  int wid     = blockIdx.x * 8 + (threadIdx.x >> 5);
  int stripsN = N >> 6;                  // N / 64
  int total   = (M >> 4) * stripsN;
  if (wid >= total) return;
  int tm    = wid / stripsN;
  int nbase = (wid - tm * stripsN) << 6;
  int lane  = threadIdx.x & 31;
  int col   = lane & 15;
  int half  = lane >> 4;

  v8f acc0, acc1, acc2, acc3;
  {
    float b0 = bias[nbase + col];
    float b1 = bias[nbase + 16 + col];
    float b2 = bias[nbase + 32 + col];
    float b3 = bias[nbase + 48 + col];
    #pragma unroll
    for (int j = 0; j < 8; ++j) { acc0[j] = b0; acc1[j] = b1; acc2[j] = b2; acc3[j] = b3; }
  }

  const float* Ap = A  + (size_t)(tm * 16 + col) * Kd + half * 2;
  const float* Bp = Bw + (size_t)(half * 2) * N + nbase + col;

  for (int k = 0; k < Kd; k += 4) {
    v2f a;
    a[0] = Ap[k];
    a[1] = Ap[k + 1];
    const float* br0 = Bp + (size_t)k * N;        // row k + half*2
    const float* br1 = Bp + (size_t)(k + 1) * N;  // row k+1 + half*2
    v2f bf0, bf1, bf2, bf3;
    bf0[0] = br0[0];  bf0[1] = br1[0];
    bf1[0] = br0[16]; bf1[1] = br1[16];
    bf2[0] = br0[32]; bf2[1] = br1[32];
    bf3[0] = br0[48]; bf3[1] = br1[48];
    acc0 = __builtin_amdgcn_wmma_f32_16x16x4_f32(false, a, false, bf0, (short)0, acc0, false, false);
    acc1 = __builtin_amdgcn_wmma_f32_16x16x4_f32(false, a, false, bf1, (short)0, acc1, false, false);
    acc2 = __builtin_amdgcn_wmma_f32_16x16x4_f32(false, a, false, bf2, (short)0, acc2, false, false);
    acc3 = __builtin_amdgcn_wmma_f32_16x16x4_f32(false, a, false, bf3, (short)0, acc3, false, false);
  }

  #pragma unroll
  for (int t = 0; t < 4; ++t) {
    v8f acc = (t == 0) ? acc0 : (t == 1) ? acc1 : (t == 2) ? acc2 : acc3;
    int colG = nbase + t * 16 + col;
    #pragma unroll
    for (int j = 0; j < 8; ++j) {
      int row = tm * 16 + half * 8 + j;
      float v = acc[j] * alpha;
      if (act) v = 0.5f * v * (1.0f + erff(v * 0.70710678118654752f));  // exact GELU
      if (resid) v += resid[(size_t)row * N + colG];
      D[(size_t)row * N + colG] = v;
    }
  }
}

// ---------------------------------------------------------------------------
// Neighborhood attention: one wave32 per (token, head); lane = channel d.
// 49 neighbors computed on the fly from the clamped 7x7 window; scores held
// in 2 per-lane slots, softmax via wave reductions, V-weighted sum per lane.
// ---------------------------------------------------------------------------
__global__ __launch_bounds__(256) void nat_attn(const float* __restrict__ Q,
                                                const float* __restrict__ KV,
                                                const float* __restrict__ rpb,
                                                float* __restrict__ Out) {
  int wid = blockIdx.x * 8 + (threadIdx.x >> 5);
  if (wid >= Mtok * NHh) return;
  int h    = wid & (NHh - 1);
  int tok  = wid >> 2;
  int lane = threadIdx.x & 31;

  int j    = tok & (Ww - 1);
  int i    = (tok >> 6) & (Hh - 1);
  int bidx = tok >> 12;

  int i0 = i - Kk / 2; if (i0 < 0) i0 = 0; if (i0 > Hh - Kk) i0 = Hh - Kk;
  int j0 = j - Kk / 2; if (j0 < 0) j0 = 0; if (j0 > Ww - Kk) j0 = Ww - Kk;

  float qd = Q[(size_t)tok * Cc + h * HDd + lane];
  const float* rp = rpb + h * (2 * Kk - 1) * (2 * Kk - 1);

  float s0 = -1e30f, s1 = -1e30f;
  for (int n = 0; n < Kk * Kk; ++n) {
    int di = n / Kk, dj = n - di * Kk;
    int nt = ((bidx * Hh + (i0 + di)) << 6) + (j0 + dj);
    float kval = KV[(size_t)nt * (2 * Cc) + h * HDd + lane];
    float p = qd * kval;
    #pragma unroll
    for (int off = 16; off; off >>= 1) p += __shfl_xor(p, off, 32);
    p += rp[(i0 + di - i + Kk - 1) * (2 * Kk - 1) + (j0 + dj - j + Kk - 1)];
    if (n < 32) { if (lane == n)      s0 = p; }
    else        { if (lane == n - 32) s1 = p; }
  }

  float mx = fmaxf(s0, s1);
  #pragma unroll
  for (int off = 16; off; off >>= 1) mx = fmaxf(mx, __shfl_xor(mx, off, 32));
  float e0 = expf(s0 - mx);
  float e1 = expf(s1 - mx);
  float sum = e0 + e1;
  #pragma unroll
  for (int off = 16; off; off >>= 1) sum += __shfl_xor(sum, off, 32);
  float inv = 1.0f / sum;

  float acc = 0.0f;
  for (int n = 0; n < Kk * Kk; ++n) {
    int di = n / Kk, dj = n - di * Kk;
    int nt = ((bidx * Hh + (i0 + di)) << 6) + (j0 + dj);
    float wn = __shfl(n < 32 ? e0 : e1, n & 31, 32);
    float vval = KV[(size_t)nt * (2 * Cc) + Cc + h * HDd + lane];
    acc += wn * vval;
  }
  Out[(size_t)tok * Cc + h * HDd + lane] = acc * inv;
}

// ---------------------------------------------------------------------------
// Orchestration.  Inputs (setup_inputs order):
//  0 query 1 key_value 2 g1 3 b1 4 g2 5 b2 6 g3 7 b3 8 Wq 9 bq 10 Wkv 11 bkv
//  12 Wp 13 bp 14 rpb 15 W1 16 bm1 17 W2 18 bm2
// ---------------------------------------------------------------------------
extern "C" void kernel_launch(void* const* d_in, const int* in_sizes, int n_in,
                              void* d_out, int out_size, void* d_ws, size_t ws_size,
                              hipStream_t stream) {
  const float* query     = (const float*)d_in[0];
  const float* key_value = (const float*)d_in[1];
  const float* g1 = (const float*)d_in[2];
  const float* b1 = (const float*)d_in[3];
  const float* g2 = (const float*)d_in[4];
  const float* b2 = (const float*)d_in[5];
  const float* g3 = (const float*)d_in[6];
  const float* b3 = (const float*)d_in[7];
  const float* Wq  = (const float*)d_in[8];
  const float* bq  = (const float*)d_in[9];
  const float* Wkv = (const float*)d_in[10];
  const float* bkv = (const float*)d_in[11];
  const float* Wp  = (const float*)d_in[12];
  const float* bp  = (const float*)d_in[13];
  const float* rpb = (const float*)d_in[14];
  const float* W1  = (const float*)d_in[15];
  const float* bm1 = (const float*)d_in[16];
  const float* W2  = (const float*)d_in[17];
  const float* bm2 = (const float*)d_in[18];
  float* out = (float*)d_out;
  float* ws  = (float*)d_ws;

  const size_t MC = (size_t)Mtok * Cc;       // 2,097,152 floats
  float* qn   = ws;                          // [0, MC)
  float* kvn  = ws + MC;                     // [MC, 2MC)
  float* Qb   = ws + 2 * MC;                 // [2MC, 3MC)
  float* KVb  = ws + 3 * MC;                 // [3MC, 5MC)  (M x 256)
  float* att  = ws;                          // reuse qn region (qn dead)
  float* X    = ws + 5 * MC;                 // [5MC, 6MC)
  float* hn   = ws + MC;                     // reuse kvn region (kvn dead)
  float* mlph = ws + 6 * MC;                 // [6MC, 10MC)  (M x 512)
  // total ws use: 10*MC floats = 80 MiB

  dim3 blk(256);
  const float alpha_q = 0.17677669529663687f;   // HD^-0.5, HD=32

  // LN(query), LN(key_value)
  ln_kernel<<<Mtok / 8, blk, 0, stream>>>(query, g1, b1, qn, Mtok);
  ln_kernel<<<Mtok / 8, blk, 0, stream>>>(key_value, g2, b2, kvn, Mtok);

  // strips of 16x64 per wave, 8 waves per block
  int stripsC  = (Mtok / 16) * (Cc / 64);       // 2048
  int strips2C = (Mtok / 16) * (2 * Cc / 64);   // 4096
  int stripsM1 = (Mtok / 16) * (MLPD_ / 64);    // 8192

  // Q = (qn @ Wq + bq) * HD^-0.5 ; KV = kvn @ Wkv + bkv
  gemm_wmma<<<stripsC / 8,  blk, 0, stream>>>(qn,  Wq,  bq,  nullptr, Qb,  Mtok, Cc,     Cc, alpha_q, 0);
  gemm_wmma<<<strips2C / 8, blk, 0, stream>>>(kvn, Wkv, bkv, nullptr, KVb, Mtok, 2 * Cc, Cc, 1.0f,    0);

  // neighborhood attention
  nat_attn<<<(Mtok * NHh) / 8, blk, 0, stream>>>(Qb, KVb, rpb, att);

  // x = key_value + att @ Wp + bp
  gemm_wmma<<<stripsC / 8, blk, 0, stream>>>(att, Wp, bp, key_value, X, Mtok, Cc, Cc, 1.0f, 0);

  // h = gelu(LN(x) @ W1 + bm1) ; out = x + h @ W2 + bm2
  ln_kernel<<<Mtok / 8, blk, 0, stream>>>(X, g3, b3, hn, Mtok);
  gemm_wmma<<<stripsM1 / 8, blk, 0, stream>>>(hn,   W1, bm1, nullptr, mlph, Mtok, MLPD_, Cc,    1.0f, 1);
  gemm_wmma<<<stripsC / 8,  blk, 0, stream>>>(mlph, W2, bm2, X,       out,  Mtok, Cc,    MLPD_, 1.0f, 0);
}